// MambaClassifier_26946624815374
// MI455X (gfx1250) — compile-verified
//
#include <hip/hip_runtime.h>
#include <math.h>

// ---------------- model constants ----------------
static constexpr int B_  = 4;
static constexpr int L_  = 2048;
static constexpr int DM  = 256;
static constexpr int NL  = 12;
static constexpr int NS  = 16;
static constexpr int DC  = 4;
static constexpr int NC  = 2;
static constexpr int DI  = 512;              // EX * DM
static constexpr int DTR = 16;               // (DM+15)/16
static constexpr int XDW = DTR + 2 * NS;     // 48
static constexpr int TOK = B_ * L_;          // 8192

typedef __attribute__((ext_vector_type(2))) float v2f;
typedef __attribute__((ext_vector_type(8))) float v8f;

// ---------------- embedding gather ----------------
__global__ void embed_k(const float* __restrict__ emb, const int* __restrict__ ids,
                        float* __restrict__ x, int total) {
    int gid = blockIdx.x * blockDim.x + threadIdx.x;
    if (gid >= total) return;
    int d   = gid & (DM - 1);
    int tok = gid >> 8;                       // DM == 256
    x[gid] = emb[ids[tok] * DM + d];
}

// ---------------- layernorm (one block of 256 per row) ----------------
__global__ void ln_k(const float* __restrict__ x, const float* __restrict__ w,
                     const float* __restrict__ b, float* __restrict__ o) {
    __shared__ float red[DM];
    int row = blockIdx.x, tid = threadIdx.x;
    float v = x[row * DM + tid];
    red[tid] = v; __syncthreads();
    for (int s = 128; s > 0; s >>= 1) { if (tid < s) red[tid] += red[tid + s]; __syncthreads(); }
    float mean = red[0] * (1.0f / DM);
    __syncthreads();
    float dv = v - mean;
    red[tid] = dv * dv; __syncthreads();
    for (int s = 128; s > 0; s >>= 1) { if (tid < s) red[tid] += red[tid + s]; __syncthreads(); }
    float var = red[0] * (1.0f / DM);
    o[row * DM + tid] = dv * rsqrtf(var + 1e-5f) * w[tid] + b[tid];
}

// ---------------- fp32 WMMA GEMM: OUT[M,N] = epi( X[M,K(ldx)] * W[N,K]^T ) ----------------
// One wave computes a 16 x (NJ*16) output tile using V_WMMA_F32_16X16X4_F32.
// EPI: 0 = plain, 1 = softplus(v + bias[n]), 2 = residual add (C initialized from R)
// NJ (compile-time): number of 16-wide N subtiles per wave (all branches fold away,
// keeping EXEC all-1s and the inner loop fully straight-line for the scheduler).
template<int EPI>
__device__ __forceinline__ float gemm_finish(float v, const float* bias, int col) {
    if (EPI == 1) {
        float s = v + bias[col];
        return (s > 20.f) ? s : log1pf(__expf(s));
    }
    return v;
}

template<int EPI, int NJ, int K>
__global__ __launch_bounds__(256) void gemm_xwt_k(
    const float* __restrict__ X, int ldx,
    const float* __restrict__ W,          // (N, K) row-major
    const float* __restrict__ bias,       // EPI==1
    const float* __restrict__ R, int ldr, // EPI==2
    float* __restrict__ OUT, int ldo,
    int ntiles64)
{
    int wv = (int)blockIdx.x * 8 + (int)(threadIdx.x >> 5);   // one wave -> one tile
    int mt = wv / ntiles64;
    int nt = wv - mt * ntiles64;
    int m0 = mt << 4;
    int n0 = nt << 6;
    int lane = (int)(threadIdx.x & 31);
    int lr   = lane & 15;            // M row (A) / N col (B,C) within tile
    int kh   = (lane >> 4) << 1;     // K sub-offset: 0 or 2 (16x16x4 f32 layout)
    int crb  = (lane >> 4) << 3;     // C/D row base: 0 or 8

    v8f c[NJ];
    #pragma unroll
    for (int j = 0; j < NJ; ++j) c[j] = (v8f){0,0,0,0,0,0,0,0};

    if (EPI == 2) {
        #pragma unroll
        for (int r = 0; r < 8; ++r) {
            const float* rr = R + (m0 + crb + r) * ldr + n0 + lr;
            #pragma unroll
            for (int j = 0; j < NJ; ++j) c[j][r] = rr[j * 16];
        }
    }

    const float* xp = X + (m0 + lr) * ldx + kh;
    const float* wp = W + (n0 + lr) * K + kh;

    #pragma unroll 4
    for (int k0 = 0; k0 < K; k0 += 4) {
        if (K > 32) __builtin_prefetch((const void*)(wp + k0 + 32), 0, 0); // global_prefetch_b8
        v2f a; a.x = xp[k0]; a.y = xp[k0 + 1];
        #pragma unroll
        for (int j = 0; j < NJ; ++j) {
            v2f bb; bb.x = wp[j * 16 * K + k0]; bb.y = wp[j * 16 * K + k0 + 1];
            c[j] = __builtin_amdgcn_wmma_f32_16x16x4_f32(false, a, false, bb, (short)0, c[j], false, false);
        }
    }

    #pragma unroll
    for (int r = 0; r < 8; ++r) {
        float* orow = OUT + (m0 + crb + r) * ldo + n0 + lr;
        #pragma unroll
        for (int j = 0; j < NJ; ++j)
            orow[j * 16] = gemm_finish<EPI>(c[j][r], bias, n0 + j * 16 + lr);
    }
}

// ---------------- causal depthwise conv (DC=4) + bias + SiLU ----------------
__global__ void conv_silu_k(const float* __restrict__ xz, const float* __restrict__ cw,
                            const float* __restrict__ cb, float* __restrict__ u, int total) {
    int gid = blockIdx.x * blockDim.x + threadIdx.x;
    if (gid >= total) return;
    int d  = gid & (DI - 1);
    int bt = gid >> 9;                 // b*L + t   (DI == 512)
    int t  = bt & (L_ - 1);
    float acc = cb[d];
    const float* wr = cw + d * DC;
    #pragma unroll
    for (int j = 0; j < DC; ++j) {
        int tt = t - (DC - 1) + j;
        if (tt >= 0) acc += xz[(bt - t + tt) * (2 * DI) + d] * wr[j];
    }
    u[gid] = acc / (1.0f + __expf(-acc));   // SiLU
}

// ---------------- selective scan: lane per (b,d,n); shfl reduce over n ----------------
// Writes y in-place over u, fused with Dskip*u and SiLU(z) gating.
__global__ void scan_k(const float* __restrict__ dt, float* __restrict__ uy,
                       const float* __restrict__ xdbl, const float* __restrict__ xz,
                       const float* __restrict__ alog, const float* __restrict__ dsk) {
    int tid = blockIdx.x * blockDim.x + threadIdx.x;     // 0 .. B_*DI*NS-1
    int n = tid & 15;
    int d = (tid >> 4) & (DI - 1);
    int b = tid >> 13;                                   // 16*512 = 8192
    float A   = -__expf(alog[d * NS + n]);
    float dsv = dsk[d];
    float h = 0.f;
    for (int t = 0; t < L_; ++t) {
        int base = b * L_ + t;
        float dtv = dt[base * DI + d];
        float uv  = uy[base * DI + d];
        float Bv  = xdbl[base * XDW + DTR + n];
        float Cv  = xdbl[base * XDW + DTR + NS + n];
        h = __expf(dtv * A) * h + (dtv * uv) * Bv;
        float p = h * Cv;
        p += __shfl_xor(p, 1, 32);
        p += __shfl_xor(p, 2, 32);
        p += __shfl_xor(p, 4, 32);
        p += __shfl_xor(p, 8, 32);
        if (n == 0) {
            float z = xz[base * (2 * DI) + DI + d];
            float sz = z / (1.0f + __expf(-z));
            uy[base * DI + d] = (p + dsv * uv) * sz;
        }
    }
}

// ---------------- masked mean pool + linear head ----------------
__global__ void poolhead_k(const float* __restrict__ xn, const int* __restrict__ ids,
                           const float* __restrict__ hw, const float* __restrict__ hb,
                           float* __restrict__ out) {
    __shared__ float red[DM];
    int b = blockIdx.x, tid = threadIdx.x;
    float acc = 0.f, cnt = 0.f;
    for (int t = 0; t < L_; ++t) {
        float m = (ids[b * L_ + t] != 0) ? 1.f : 0.f;
        acc += m * xn[(b * L_ + t) * DM + tid];
        cnt += m;
    }
    float pooled = acc / fmaxf(cnt, 1.f);
    for (int c = 0; c < NC; ++c) {
        red[tid] = pooled * hw[c * DM + tid];
        __syncthreads();
        for (int s = 128; s > 0; s >>= 1) { if (tid < s) red[tid] += red[tid + s]; __syncthreads(); }
        if (tid == 0) out[b * NC + c] = red[0] + hb[c];
        __syncthreads();
    }
}

// ---------------- host orchestration ----------------
extern "C" void kernel_launch(void* const* d_in, const int* in_sizes, int n_in,
                              void* d_out, int out_size, void* d_ws, size_t ws_size,
                              hipStream_t stream) {
    (void)in_sizes; (void)n_in; (void)out_size; (void)ws_size;
    const float* emb  = (const float*)d_in[0];
    const float* lnw  = (const float*)d_in[1];
    const float* lnb  = (const float*)d_in[2];
    const float* ipw  = (const float*)d_in[3];
    const float* cw   = (const float*)d_in[4];
    const float* cb   = (const float*)d_in[5];
    const float* xpw  = (const float*)d_in[6];
    const float* dpw  = (const float*)d_in[7];
    const float* dpb  = (const float*)d_in[8];
    const float* alog = (const float*)d_in[9];
    const float* dsk  = (const float*)d_in[10];
    const float* opw  = (const float*)d_in[11];
    const float* flnw = (const float*)d_in[12];
    const float* flnb = (const float*)d_in[13];
    const float* hw   = (const float*)d_in[14];
    const float* hb   = (const float*)d_in[15];
    const int*   ids  = (const int*)d_in[16];
    float* out = (float*)d_out;

    // workspace partition (fp32): x, xn, xz, u(=y), xdbl, dt  ~= 85.5 MB
    float* x    = (float*)d_ws;
    float* xn   = x    + TOK * DM;
    float* xz   = xn   + TOK * DM;
    float* u    = xz   + TOK * 2 * DI;
    float* xdbl = u    + TOK * DI;
    float* dt   = xdbl + TOK * XDW;

    embed_k<<<(TOK * DM) / 256, 256, 0, stream>>>(emb, ids, x, TOK * DM);

    for (int l = 0; l < NL; ++l) {
        ln_k<<<TOK, 256, 0, stream>>>(x, lnw + l * DM, lnb + l * DM, xn);

        {   // xz = xn @ ipw^T : (8192,256) x (1024,256)^T ; NJ=4, K=256
            int nt64 = (2 * DI) / 64, tiles = (TOK / 16) * nt64;         // 8192 tiles
            gemm_xwt_k<0, 4, DM><<<tiles / 8, 256, 0, stream>>>(
                xn, DM, ipw + l * 2 * DI * DM, nullptr, nullptr, 0, xz, 2 * DI, nt64);
        }

        conv_silu_k<<<(TOK * DI) / 256, 256, 0, stream>>>(xz, cw + l * DI * DC, cb + l * DI, u, TOK * DI);

        {   // xdbl = u @ xpw^T : (8192,512) x (48,512)^T ; NJ=3 (48 cols), K=512
            int nt64 = 1, tiles = (TOK / 16) * nt64;                     // 512 tiles
            gemm_xwt_k<0, 3, DI><<<tiles / 8, 256, 0, stream>>>(
                u, DI, xpw + l * XDW * DI, nullptr, nullptr, 0, xdbl, XDW, nt64);
        }

        {   // dt = softplus(xdbl[:, :16] @ dpw^T + dpb) ; NJ=4, K=16 (fully unrolled)
            int nt64 = DI / 64, tiles = (TOK / 16) * nt64;               // 4096 tiles
            gemm_xwt_k<1, 4, DTR><<<tiles / 8, 256, 0, stream>>>(
                xdbl, XDW, dpw + l * DI * DTR, dpb + l * DI, nullptr, 0, dt, DI, nt64);
        }

        scan_k<<<(B_ * DI * NS) / 256, 256, 0, stream>>>(
            dt, u, xdbl, xz, alog + l * DI * NS, dsk + l * DI);

        {   // x = x + y @ opw^T : (8192,512) x (256,512)^T ; residual fused, NJ=4, K=512
            int nt64 = DM / 64, tiles = (TOK / 16) * nt64;               // 2048 tiles
            gemm_xwt_k<2, 4, DI><<<tiles / 8, 256, 0, stream>>>(
                u, DI, opw + l * DM * DI, nullptr, x, DM, x, DM, nt64);
        }
    }

    ln_k<<<TOK, 256, 0, stream>>>(x, flnw, flnb, xn);
    poolhead_k<<<B_, 256, 0, stream>>>(xn, ids, hw, hb, out);
}